// AreaLoss_82025285419449
// MI455X (gfx1250) — compile-verified
//
#include <hip/hip_runtime.h>

typedef __attribute__((ext_vector_type(2))) float v2f;
typedef __attribute__((ext_vector_type(4))) float v4f;
typedef __attribute__((ext_vector_type(8))) float v8f;

#define B_DIM   64
#define C_DIM   1000
#define HW      784          // 28*28
#define K_TOP   25
#define NPAIRS  (B_DIM * K_TOP)       // 1600
#define N_P     (B_DIM * HW)          // 50176 = 392 chunks of 128
#define INV_DEN (1.0f / 50176.0f)
#define NEG_BIG (-3.402823466e+38f)

// --- WMMA-based accumulators ---------------------------------------------
// A = data tile (16x4 f32, 2 VGPRs/lane), B = ones (4x16).
// D[i][j] += sum_k A[i][k]  => every A element lands in exactly 16 of the
// 256 D elements, so sum(D) == 16 * sum(all accumulated A tiles). Exact f32.
__device__ __forceinline__ v8f wmma_sum64(v2f a, v8f c) {
    v2f ones;
    ones[0] = 1.0f;
    ones[1] = 1.0f;
    return __builtin_amdgcn_wmma_f32_16x16x4_f32(
        /*neg_a=*/false, a, /*neg_b=*/false, ones,
        /*c_mod=*/(short)0, c, /*reuse_a=*/false, /*reuse_b=*/false);
}

__device__ __forceinline__ v8f wmma_sum128(v4f a, v8f c) {
    v2f lo, hi;
    lo[0] = a[0]; lo[1] = a[1];
    hi[0] = a[2]; hi[1] = a[3];
    c = wmma_sum64(lo, c);
    c = wmma_sum64(hi, c);
    return c;
}

__device__ __forceinline__ float wave_reduce_c(v8f c) {
    float s = c[0] + c[1] + c[2] + c[3] + c[4] + c[5] + c[6] + c[7];
    #pragma unroll
    for (int off = 16; off > 0; off >>= 1)
        s += __shfl_down(s, off, 32);
    return s;   // valid in lane 0
}

// Wave-uniform (SGPR) wave id: keeps reduction loops scalar so EXEC is
// provably all-1s at every WMMA (ISA requirement).
__device__ __forceinline__ int uniform_wave_id() {
    int w = (int)((blockIdx.x * blockDim.x + threadIdx.x) >> 5);
    return __builtin_amdgcn_readfirstlane(w);
}

// --- Kernel 1: per-batch top-25 indices of main_out row (softmax monotone)
__global__ void topk_kernel(const float* __restrict__ main_out,
                            int* __restrict__ idx_out,
                            float* __restrict__ out) {
    __shared__ float vals[C_DIM];
    __shared__ float redv[256];
    __shared__ int   redi[256];
    const int b = blockIdx.x;
    const int t = threadIdx.x;

    if (b == 0 && t == 0) out[0] = 0.0f;   // zero the poisoned output

    for (int i = t; i < C_DIM; i += 256)
        vals[i] = main_out[b * C_DIM + i];
    __syncthreads();

    for (int k = 0; k < K_TOP; ++k) {
        float best = NEG_BIG;
        int   bi   = C_DIM;
        for (int i = t; i < C_DIM; i += 256) {
            float v = vals[i];
            if (v > best) { best = v; bi = i; }
        }
        redv[t] = best; redi[t] = bi;
        __syncthreads();
        #pragma unroll
        for (int s = 128; s > 0; s >>= 1) {
            if (t < s) {
                float ov = redv[t + s]; int oi = redi[t + s];
                if (ov > redv[t] || (ov == redv[t] && oi < redi[t])) {
                    redv[t] = ov; redi[t] = oi;
                }
            }
            __syncthreads();
        }
        if (t == 0) {
            idx_out[b * K_TOP + k] = redi[0];
            vals[redi[0]] = NEG_BIG;           // exclude from later rounds
        }
        __syncthreads();
    }
}

// --- Kernel 2: sum(p) / denom: b128 loads + WMMA reduction ----------------
__global__ void sum_p_kernel(const float* __restrict__ p,
                             float* __restrict__ out) {
    const int lane   = threadIdx.x & 31;
    const int gwave  = uniform_wave_id();
    const int nwaves = (int)((gridDim.x * blockDim.x) >> 5);

    v8f c = {};
    for (int chunk = gwave; chunk < N_P / 128; chunk += nwaves) {
        v4f a = *(const v4f*)(p + chunk * 128 + lane * 4);   // global_load_b128
        c = wmma_sum128(a, c);
    }
    float s = wave_reduce_c(c);
    if (lane == 0)
        atomicAdd(out, s * (INV_DEN * (1.0f / 16.0f)));
}

// --- Kernel 3: sum of gathered feature slabs / denom ----------------------
__global__ void gather_sum_kernel(const float* __restrict__ features,
                                  const int* __restrict__ idx,
                                  float* __restrict__ out) {
    const int lane   = threadIdx.x & 31;
    const int gwave  = uniform_wave_id();
    const int nwaves = (int)((gridDim.x * blockDim.x) >> 5);

    v8f c = {};
    for (int pair = gwave; pair < NPAIRS; pair += nwaves) {
        const int b  = pair / K_TOP;
        const int ch = idx[pair];
        const float* base = features + ((size_t)b * C_DIM + (size_t)ch) * HW;

        // 784 = 6*128 + 16 : six b128 chunks, one lane-masked b64 tail
        #pragma unroll
        for (int chunk = 0; chunk < 6; ++chunk) {
            v4f a = *(const v4f*)(base + chunk * 128 + lane * 4);
            c = wmma_sum128(a, c);
        }
        {
            const int off = 6 * 128 + lane * 2;
            v2f a;
            if (off < HW) {                 // lanes 0..7 only
                a = *(const v2f*)(base + off);
            } else {
                a[0] = 0.0f; a[1] = 0.0f;
            }
            c = wmma_sum64(a, c);           // full EXEC here (wave-uniform)
        }
    }
    float s = wave_reduce_c(c);
    if (lane == 0)
        atomicAdd(out, s * (INV_DEN * (1.0f / 16.0f)));
}

extern "C" void kernel_launch(void* const* d_in, const int* in_sizes, int n_in,
                              void* d_out, int out_size, void* d_ws, size_t ws_size,
                              hipStream_t stream) {
    const float* p        = (const float*)d_in[0];   // [64,1,28,28]
    const float* main_out = (const float*)d_in[1];   // [64,1000]
    const float* features = (const float*)d_in[2];   // [64,1000,28,28]
    float* out = (float*)d_out;                      // scalar
    int*   idx = (int*)d_ws;                         // 1600 ints of scratch

    (void)in_sizes; (void)n_in; (void)out_size; (void)ws_size;

    // Kernel order on one stream => serialized: zero-out happens in topk
    // before any atomicAdd in the later kernels.
    topk_kernel<<<B_DIM, 256, 0, stream>>>(main_out, idx, out);
    sum_p_kernel<<<16, 256, 0, stream>>>(p, out);             // 128 waves
    gather_sum_kernel<<<200, 256, 0, stream>>>(features, idx, out); // 1600 waves
}